// _NetworkGNNNet_6279242187301
// MI455X (gfx1250) — compile-verified
//
#include <hip/hip_runtime.h>
#include <math.h>

typedef float v2f __attribute__((ext_vector_type(2)));
typedef float v8f __attribute__((ext_vector_type(8)));

// ---------------- elementwise / edge kernels ----------------

__global__ void k_init_deg(float* __restrict__ deg, int n) {
    int i = blockIdx.x * blockDim.x + threadIdx.x;
    if (i < n) deg[i] = 1.0f;  // self-loop contributes 1 to every node's degree
}

__global__ void k_deg_edges(const int* __restrict__ dst, float* __restrict__ deg, int e) {
    int i = blockIdx.x * blockDim.x + threadIdx.x;
    if (i < e) atomicAdd(&deg[dst[i]], 1.0f);
}

// deg -> dinv (in place) and initialize xa with self-loop term dinv^2 * x
__global__ void k_dinv_selfx(const float* __restrict__ x, float* __restrict__ deg_dinv,
                             float* __restrict__ xa, int n) {
    int i = blockIdx.x * blockDim.x + threadIdx.x;
    if (i < n) {
        float d  = deg_dinv[i];
        float di = d > 0.0f ? rsqrtf(d) : 0.0f;
        deg_dinv[i] = di;
        float d2 = di * di;
        float4 xs = ((const float4*)x)[i];
        float4 o;
        o.x = d2 * xs.x; o.y = d2 * xs.y; o.z = d2 * xs.z; o.w = d2 * xs.w;
        ((float4*)xa)[i] = o;
    }
}

// layer-1 aggregation on RAW features (4 floats/edge): xa[dst] += norm * x[src]
__global__ void k_edge_agg4(const int* __restrict__ src, const int* __restrict__ dst,
                            const float* __restrict__ dinv, const float* __restrict__ x,
                            float* __restrict__ xa, int e) {
    int i = blockIdx.x * blockDim.x + threadIdx.x;
    if (i < e) {
        int s = src[i], d = dst[i];
        float nrm = dinv[s] * dinv[d];
        float4 xs = ((const float4*)x)[s];
        atomicAdd(&xa[4 * d + 0], nrm * xs.x);
        atomicAdd(&xa[4 * d + 1], nrm * xs.y);
        atomicAdd(&xa[4 * d + 2], nrm * xs.z);
        atomicAdd(&xa[4 * d + 3], nrm * xs.w);
    }
}

// ---------------- WMMA node MLP: z = relu(xa @ W1 + b1) @ W2 ----------------
// One wave32 per 16-node tile. A = xa tile (16x4 f32), B = W1 half (4x16 f32),
// D = hidden (16x16 f32). Two WMMAs cover all 32 hidden channels.
__global__ void k_node_mlp_wmma(const float* __restrict__ xa,
                                const float* __restrict__ W1,  // [4][32] row-major
                                const float* __restrict__ b1,  // [32]
                                const float* __restrict__ W2,  // [32]
                                float* __restrict__ z, int n) {
    int lane = threadIdx.x & 31;
    int wave = threadIdx.x >> 5;
    int tile = blockIdx.x * (blockDim.x >> 5) + wave;
    int base = tile << 4;
    int half = lane >> 4;   // 0: lanes 0-15 (K=0,1 / M rows 0-7), 1: lanes 16-31
    int l    = lane & 15;

    // A-matrix 16x4 f32 layout: lane half selects K pair {0,1} or {2,3}; M = l
    int node  = base + l;
    bool valid = node < n;
    int nc = valid ? node : (n - 1);
    v2f a;
    a.x = valid ? xa[nc * 4 + half * 2 + 0] : 0.0f;
    a.y = valid ? xa[nc * 4 + half * 2 + 1] : 0.0f;

    // B-matrix 4x16 f32: rows K = {0,1} / {2,3} per half, column = l (+16 for hi tile)
    int r = half * 2;
    v2f blo, bhi;
    blo.x = W1[(r + 0) * 32 + l];
    blo.y = W1[(r + 1) * 32 + l];
    bhi.x = W1[(r + 0) * 32 + 16 + l];
    bhi.y = W1[(r + 1) * 32 + 16 + l];

    v8f c0 = {0.f, 0.f, 0.f, 0.f, 0.f, 0.f, 0.f, 0.f};
    v8f c1 = {0.f, 0.f, 0.f, 0.f, 0.f, 0.f, 0.f, 0.f};
    // args: (neg_a, A, neg_b, B, c_mod, C, reuse_a, reuse_b)
    c0 = __builtin_amdgcn_wmma_f32_16x16x4_f32(false, a, false, blo, (short)0, c0, false, false);
    c1 = __builtin_amdgcn_wmma_f32_16x16x4_f32(false, a, false, bhi, (short)0, c1, false, false);

    // This lane holds hidden columns (l) and (16+l) for nodes M = j + 8*half.
    float b1lo = b1[l],  b1hi = b1[16 + l];
    float w2lo = W2[l],  w2hi = W2[16 + l];

    float p[8];
#pragma unroll
    for (int j = 0; j < 8; ++j) {
        float h0 = c0[j] + b1lo; h0 = h0 > 0.0f ? h0 : 0.0f;
        float h1 = c1[j] + b1hi; h1 = h1 > 0.0f ? h1 : 0.0f;
        p[j] = h0 * w2lo + h1 * w2hi;
    }
    // Reduce over the 16 lanes of this half (xor masks 1..8 stay within the half)
#pragma unroll
    for (int j = 0; j < 8; ++j) {
        float v = p[j];
        v += __shfl_xor(v, 1, 32);
        v += __shfl_xor(v, 2, 32);
        v += __shfl_xor(v, 4, 32);
        v += __shfl_xor(v, 8, 32);
        p[j] = v;
    }
    if (l == 0) {
#pragma unroll
        for (int j = 0; j < 8; ++j) {
            int m = base + half * 8 + j;   // lane0 -> M=0..7, lane16 -> M=8..15
            if (m < n) z[m] = p[j];
        }
    }
}

// ---------------- layer-2 scalar aggregation + sigmoid ----------------

__global__ void k_init_acc(const float* __restrict__ dinv, const float* __restrict__ z,
                           float* __restrict__ acc, int n) {
    int i = blockIdx.x * blockDim.x + threadIdx.x;
    if (i < n) {
        float di = dinv[i];
        acc[i] = di * di * z[i];   // self-loop term
    }
}

__global__ void k_edge_agg1(const int* __restrict__ src, const int* __restrict__ dst,
                            const float* __restrict__ dinv, const float* __restrict__ z,
                            float* __restrict__ acc, int e) {
    int i = blockIdx.x * blockDim.x + threadIdx.x;
    if (i < e) {
        int s = src[i], d = dst[i];
        atomicAdd(&acc[d], dinv[s] * dinv[d] * z[s]);
    }
}

__global__ void k_final(const float* __restrict__ acc, const float* __restrict__ b2,
                        float* __restrict__ out, int n) {
    int i = blockIdx.x * blockDim.x + threadIdx.x;
    if (i < n) {
        float t = acc[i] + b2[0];
        out[i] = 1.0f / (1.0f + expf(-t));
    }
}

// ---------------- launcher ----------------

extern "C" void kernel_launch(void* const* d_in, const int* in_sizes, int n_in,
                              void* d_out, int out_size, void* d_ws, size_t ws_size,
                              hipStream_t stream) {
    const float* x  = (const float*)d_in[0];   // [N,4]
    const int*   ei = (const int*)d_in[1];     // [2,E]
    const float* W1 = (const float*)d_in[2];   // [4,32]
    const float* b1 = (const float*)d_in[3];   // [32]
    const float* W2 = (const float*)d_in[4];   // [32,1]
    const float* b2 = (const float*)d_in[5];   // [1]

    int n = in_sizes[0] / 4;
    int e = in_sizes[1] / 2;
    const int* src = ei;
    const int* dst = ei + e;

    // Workspace layout (floats): [deg/dinv: n][xa: 4n][z: n][acc: n] = 7n floats
    float* ws   = (float*)d_ws;
    float* dinv = ws;                       // n (deg, then dinv in place)
    float* xa   = ws + (size_t)n;           // 4n, 16B aligned (n*4B % 16 == 0)
    float* zbuf = ws + (size_t)5 * n;       // n
    float* acc  = ws + (size_t)6 * n;       // n
    float* out  = (float*)d_out;

    const int TB = 256;
    int gn = (n + TB - 1) / TB;
    int ge = (e + TB - 1) / TB;
    int gw = (n + 127) / 128;               // 8 waves/block, 16 nodes/wave

    k_init_deg     <<<gn, TB, 0, stream>>>(dinv, n);
    k_deg_edges    <<<ge, TB, 0, stream>>>(dst, dinv, e);
    k_dinv_selfx   <<<gn, TB, 0, stream>>>(x, dinv, xa, n);
    k_edge_agg4    <<<ge, TB, 0, stream>>>(src, dst, dinv, x, xa, e);
    k_node_mlp_wmma<<<gw, TB, 0, stream>>>(xa, W1, b1, W2, zbuf, n);
    k_init_acc     <<<gn, TB, 0, stream>>>(dinv, zbuf, acc, n);
    k_edge_agg1    <<<ge, TB, 0, stream>>>(src, dst, dinv, zbuf, acc, e);
    k_final        <<<gn, TB, 0, stream>>>(acc, b2, out, n);
}